// StructNConv2D_d_with_g_34505767256406
// MI455X (gfx1250) — compile-verified
//
#include <hip/hip_runtime.h>
#include <hip/hip_bf16.h>
#include <cstdint>
#include <cstddef>

// ---------------- problem constants ----------------
#define EPSF 1e-20f
#define CHN  32
#define HDIM 256
#define WDIM 256
#define HWSZ (HDIM * WDIM)          // 65536
#define NBAT 2
#define ELEMS (NBAT * CHN * HWSZ)   // 4194304 per tensor

// ---------------- workspace layout (floats) ----------------
// [0,288)      softplus(spatial_weight)   (C_IN x 9)
// [288,1312)   softplus(channel_weight)   (C_OUT x C_IN)
// [1312,1344)  softplus(w_grad)           (C_IN)
// [1344]       sum(softplus(sw)),  [1345] sum(softplus(cw))
// [2048, 2048+ELEMS)         ds  (d_spatial)
// [2048+ELEMS, 2048+2*ELEMS) cds (cd_spatial)
#define WS_SW   0
#define WS_CW   288
#define WS_WG   1312
#define WS_SUM  1344
#define WS_DS   2048
#define WS_CDS  (2048 + ELEMS)

typedef __attribute__((ext_vector_type(2))) float v2f;
typedef __attribute__((ext_vector_type(8))) float v8f;

__device__ __forceinline__ float softplus_f(float x) {
  return (x > 0.f) ? (x + log1pf(expf(-x))) : log1pf(expf(x));
}

// ---------------- kernel 0: softplus weights + scalar sums ----------------
__global__ __launch_bounds__(512)
void prep_kernel(const float* __restrict__ w_grad,
                 const float* __restrict__ spatial_weight,
                 const float* __restrict__ channel_weight,
                 float* __restrict__ wsf) {
  int t = threadIdx.x;
  for (int i = t; i < 288; i += 512)  wsf[WS_SW + i] = softplus_f(spatial_weight[i]);
  for (int i = t; i < 1024; i += 512) wsf[WS_CW + i] = softplus_f(channel_weight[i]);
  if (t < 32)                         wsf[WS_WG + t] = softplus_f(w_grad[t]);
  __syncthreads();
  if (t == 0) {
    float s = 0.f;
    for (int i = 0; i < 288; ++i) s += wsf[WS_SW + i];
    wsf[WS_SUM + 0] = s;
    s = 0.f;
    for (int i = 0; i < 1024; ++i) s += wsf[WS_CW + i];
    wsf[WS_SUM + 1] = s;
  }
}

// ---------------- kernel 1: 3x3 confidence-propagation stencil ----------------
// One thread per (b,c,h,w). Neighbors come out of L1/L2; pixel axis is
// contiguous so all 6 tensor streams are coalesced.
__global__ __launch_bounds__(256)
void spatial_kernel(const float* __restrict__ d,  const float* __restrict__ cd,
                    const float* __restrict__ gx, const float* __restrict__ cgx,
                    const float* __restrict__ gy, const float* __restrict__ cgy,
                    const float* __restrict__ wsf,
                    float* __restrict__ ds_buf, float* __restrict__ cds_buf) {
  int idx = blockIdx.x * blockDim.x + threadIdx.x;     // [0, ELEMS)
  int w = idx & (WDIM - 1);
  int h = (idx >> 8) & (HDIM - 1);
  int c = (idx >> 16) & (CHN - 1);

  float wg      = wsf[WS_WG + c];
  float inv1pwg = 1.f / (1.f + wg);
  float inv_ssw = 1.f / wsf[WS_SUM + 0];

  float gx0 = gx[idx], cgx0 = cgx[idx];
  float gy0 = gy[idx], cgy0 = cgy[idx];

  float nom = 0.f, den = 0.f;
#pragma unroll
  for (int r = 0; r < 3; ++r) {
#pragma unroll
    for (int cc = 0; cc < 3; ++cc) {
      int j  = r * 3 + cc;
      int dy = r - 1, dx = cc - 1;
      int hh = h + dy, wn = w + dx;
      bool inb = ((unsigned)hh < (unsigned)HDIM) && ((unsigned)wn < (unsigned)WDIM);
      int nidx = idx + dy * WDIM + dx;
      float dr = 0.f, cdr = 0.f, gxr = 0.f, cgxr = 0.f, gyr = 0.f, cgyr = 0.f;
      if (inb) {
        dr = d[nidx];  cdr = cd[nidx];
        gxr = gx[nidx]; cgxr = cgx[nidx];
        gyr = gy[nidx]; cgyr = cgy[nidx];
      }
      float gx_prop  = (cgxr * gxr + cgx0 * gx0) / (cgxr + cgx0 + EPSF);
      float cgx_prop = 0.5f * (cgxr + cgx0);
      float gy_prop  = (cgyr * gyr + cgy0 * gy0) / (cgyr + cgy0 + EPSF);
      float cgy_prop = 0.5f * (cgyr + cgy0);
      float fdx = (float)(-dx), fdy = (float)(-dy);      // dist1 = [1,0,-1]
      float adx = fabsf(fdx),   ady = fabsf(fdy);
      float g_prop  = fdx * gx_prop + fdy * gy_prop;
      float cg_prop = (adx * cgx_prop + ady * cgy_prop) / (adx + ady + EPSF);
      float d_prop  = dr * (1.f + g_prop);
      float cd_prop = cdr * (1.f + wg * cg_prop) * inv1pwg;
      float sw = wsf[WS_SW + c * 9 + j];
      nom += cd_prop * d_prop * sw;
      den += cd_prop * sw;
    }
  }
  ds_buf[idx]  = nom / (den + EPSF);
  cds_buf[idx] = den * inv_ssw;
}

// ---------------- kernel 2: channel mixing via V_WMMA_F32_16X16X4_F32 ----------
// D[o][p] = sum_i A[o][i] * B[i][p]  with A = softplus(cw) (32x32, preloaded),
// B = {cds*ds , cds} for a 16-pixel tile. M = out-channels (2 tiles of 16),
// K = in-channels (8 steps of 4), N = 16 pixels -> coalesced loads & stores.
__global__ __launch_bounds__(256)
void channel_wmma_kernel(const float* __restrict__ wsf,
                         const float* __restrict__ bias,
                         float* __restrict__ out) {
  const float* __restrict__ cw   = wsf + WS_CW;
  const float* __restrict__ dsb  = wsf + WS_DS;
  const float* __restrict__ cdsb = wsf + WS_CDS;
  float inv_scw = 1.f / wsf[WS_SUM + 1];

  int lane = threadIdx.x & 31;
  int l    = lane & 15;
  int hi   = (lane >> 4) & 1;          // 0: lanes 0-15, 1: lanes 16-31
  int wave = (blockIdx.x * blockDim.x + threadIdx.x) >> 5;
  int nwaves = (gridDim.x * blockDim.x) >> 5;

  // Preload A fragments: a[t][s] covers M = 16t..16t+15, K = 4s..4s+3.
  // ISA layout: lanes 0-15 hold (K=4s, 4s+1), lanes 16-31 hold (K=4s+2, 4s+3).
  v2f a[2][8];
#pragma unroll
  for (int t = 0; t < 2; ++t) {
#pragma unroll
    for (int s = 0; s < 8; ++s) {
      int kb  = 4 * s + 2 * hi;
      int row = 16 * t + l;
      v2f av;
      av.x = cw[row * CHN + kb];
      av.y = cw[row * CHN + kb + 1];
      a[t][s] = av;
    }
  }
  // Preload bias per (tile, acc-row): D VGPR r holds M = r (lanes<16) / r+8.
  float bfrag[2][8];
#pragma unroll
  for (int t = 0; t < 2; ++t)
#pragma unroll
    for (int r = 0; r < 8; ++r)
      bfrag[t][r] = bias[16 * t + r + 8 * hi];

  const int NTILES = NBAT * HWSZ / 16;      // 8192 pixel tiles
  for (int tile = wave; tile < NTILES; tile += nwaves) {
    int b   = tile >> 12;                   // 4096 tiles per batch
    int pix = (tile & 4095) << 4;           // pixel base within batch

    v8f accN0 = {0.f,0.f,0.f,0.f,0.f,0.f,0.f,0.f};
    v8f accN1 = {0.f,0.f,0.f,0.f,0.f,0.f,0.f,0.f};
    v8f accD0 = {0.f,0.f,0.f,0.f,0.f,0.f,0.f,0.f};
    v8f accD1 = {0.f,0.f,0.f,0.f,0.f,0.f,0.f,0.f};

#pragma unroll
    for (int s = 0; s < 8; ++s) {
      int kb = 4 * s + 2 * hi;              // B rows this half-wave supplies
      size_t base = ((size_t)(b * CHN + kb)) * HWSZ + (size_t)pix + (size_t)l;
      float c0 = cdsb[base];
      float c1 = cdsb[base + HWSZ];
      float d0 = dsb[base];
      float d1 = dsb[base + HWSZ];
      v2f bden; bden.x = c0;      bden.y = c1;
      v2f bnom; bnom.x = c0 * d0; bnom.y = c1 * d1;
      accN0 = __builtin_amdgcn_wmma_f32_16x16x4_f32(false, a[0][s], false, bnom, (short)0, accN0, false, false);
      accN1 = __builtin_amdgcn_wmma_f32_16x16x4_f32(false, a[1][s], false, bnom, (short)0, accN1, false, false);
      accD0 = __builtin_amdgcn_wmma_f32_16x16x4_f32(false, a[0][s], false, bden, (short)0, accD0, false, false);
      accD1 = __builtin_amdgcn_wmma_f32_16x16x4_f32(false, a[1][s], false, bden, (short)0, accD1, false, false);
    }

    // Epilogue: fused divide + bias; d_out then cd_out (concatenated outputs).
#pragma unroll
    for (int r = 0; r < 8; ++r) {
      {
        int o = r + 8 * hi;
        float n = accN0[r], de = accD0[r];
        size_t addr = ((size_t)(b * CHN + o)) * HWSZ + (size_t)pix + (size_t)l;
        out[addr]                 = n / (de + EPSF) + bfrag[0][r];
        out[(size_t)ELEMS + addr] = de * inv_scw;
      }
      {
        int o = 16 + r + 8 * hi;
        float n = accN1[r], de = accD1[r];
        size_t addr = ((size_t)(b * CHN + o)) * HWSZ + (size_t)pix + (size_t)l;
        out[addr]                 = n / (de + EPSF) + bfrag[1][r];
        out[(size_t)ELEMS + addr] = de * inv_scw;
      }
    }
  }
}

// ---------------- launch ----------------
extern "C" void kernel_launch(void* const* d_in, const int* in_sizes, int n_in,
                              void* d_out, int out_size, void* d_ws, size_t ws_size,
                              hipStream_t stream) {
  const float* d    = (const float*)d_in[0];
  const float* cd   = (const float*)d_in[1];
  const float* gx   = (const float*)d_in[2];
  const float* cgx  = (const float*)d_in[3];
  const float* gy   = (const float*)d_in[4];
  const float* cgy  = (const float*)d_in[5];
  const float* wgr  = (const float*)d_in[6];  // w_grad          (32)
  const float* sw   = (const float*)d_in[7];  // spatial_weight  (288)
  const float* cw   = (const float*)d_in[8];  // channel_weight  (1024)
  const float* bias = (const float*)d_in[9];  // bias            (32)
  float* wsf = (float*)d_ws;
  float* out = (float*)d_out;

  prep_kernel<<<1, 512, 0, stream>>>(wgr, sw, cw, wsf);
  spatial_kernel<<<ELEMS / 256, 256, 0, stream>>>(d, cd, gx, cgx, gy, cgy, wsf,
                                                  wsf + WS_DS, wsf + WS_CDS);
  // 256 blocks x 256 threads = 2048 waves; 4 pixel-tiles per wave, so the
  // 32x32 weight-fragment preload amortizes over 128 WMMA issues per wave.
  channel_wmma_kernel<<<256, 256, 0, stream>>>(wsf, bias, out);
}